// sxsGAT_91164975825473
// MI455X (gfx1250) — compile-verified
//
#include <hip/hip_runtime.h>
#include <hip/hip_bf16.h>

// GAT (2 layers) + MLP head, fused flash-style softmax attention.
// K=64 graphs, N=1024 nodes, DIN=16, H1=4, DOUT=8, FC1=54.
// CDNA5 paths used:
//   - V_WMMA_F32_16X16X4_F32 (pure f32 WMMA) for att@h and the head GEMMs
//   - GLOBAL_LOAD_ASYNC_TO_LDS_B128 + s_wait_asynccnt for LDS staging

typedef __attribute__((ext_vector_type(2))) float v2f;
typedef __attribute__((ext_vector_type(8))) float v8f;

#define ALPHA 0.01f
#define NEGINF (-1.0e12f)

__device__ __forceinline__ void async_copy_b128(uint32_t lds_addr, const void* gaddr) {
    asm volatile("global_load_async_to_lds_b128 %0, %1, off"
                 :: "v"(lds_addr), "v"((unsigned long long)(uintptr_t)gaddr)
                 : "memory");
}
__device__ __forceinline__ void wait_async0() {
    asm volatile("s_wait_asynccnt 0" ::: "memory");
}

// ---------------------------------------------------------------------------
// Kernel 1: per-node projection  h = act @ W, f1 = h . a[:F], f2 = h . a[F:]
// One thread per (k, n). Tiny, memory-bound.
// ---------------------------------------------------------------------------
template <int FIN, int FOUT>
__global__ __launch_bounds__(256) void hf_kernel(
    const float* __restrict__ act,   // [K*N, FIN]
    const float* __restrict__ W,     // [FIN, FOUT]
    const float* __restrict__ a,     // [2*FOUT]
    float* __restrict__ h,           // [K*N, FOUT]
    float* __restrict__ f1,          // [K*N]
    float* __restrict__ f2,          // [K*N]
    int total)
{
    int idx = blockIdx.x * blockDim.x + threadIdx.x;
    if (idx >= total) return;
    float x[FIN];
    const float* p = act + (size_t)idx * FIN;
#pragma unroll
    for (int d = 0; d < FIN; ++d) x[d] = p[d];
    float s1 = 0.f, s2 = 0.f;
#pragma unroll
    for (int f = 0; f < FOUT; ++f) {
        float hv = 0.f;
#pragma unroll
        for (int d = 0; d < FIN; ++d) hv += x[d] * W[d * FOUT + f];
        h[(size_t)idx * FOUT + f] = hv;
        s1 += hv * a[f];
        s2 += hv * a[FOUT + f];
    }
    f1[idx] = s1;
    f2[idx] = s2;
}

// ---------------------------------------------------------------------------
// Kernel 2: fused masked-softmax attention + aggregation + ReLU.
// Grid: (N/16 row tiles, K batches).  Block: 128 threads (4 waves).
// Never materializes e[N,N]: streaming max/sum, then WMMA-accumulated
// sum_j exp(e_ij - m_i) * h_j, divided by s_i at the end.
// ---------------------------------------------------------------------------
template <int F>
__global__ __launch_bounds__(128) void gat_attn_kernel(
    const float* __restrict__ hbuf,  // [K, N, F]
    const float* __restrict__ f1g,   // [K, N]
    const float* __restrict__ f2g,   // [K, N]
    const int*   __restrict__ adj,   // [N, N]
    float* __restrict__ out)         // [K, N, F]  (ReLU applied)
{
    constexpr int N = 1024;
    const int k  = blockIdx.y;
    const int i0 = blockIdx.x * 16;
    const int t  = threadIdx.x;

    __shared__ __align__(16) float s_f2[N];
    __shared__ __align__(16) float s_h[N][F];
    __shared__ float s_f1[16];
    __shared__ __align__(4) unsigned char s_adj[16][N];
    __shared__ float s_m[16], s_s[16];
    __shared__ float s_red[16][8];
    __shared__ float s_C[4][16][16];

    // ---- async staging of f2 and h into LDS (ASYNCcnt path) ----
    {
        uint32_t lds_f2 = (uint32_t)(uintptr_t)(void*)&s_f2[0];
        const float* gf2 = f2g + k * N;
#pragma unroll
        for (int it = 0; it < (N / 4) / 128; ++it) {          // 2 x b128
            int idx = t + it * 128;
            async_copy_b128(lds_f2 + idx * 16, gf2 + idx * 4);
        }
        uint32_t lds_h = (uint32_t)(uintptr_t)(void*)&s_h[0][0];
        const float* gh = hbuf + (size_t)k * N * F;
#pragma unroll
        for (int it = 0; it < (N * F / 4) / 128; ++it) {      // 8 or 16 x b128
            int idx = t + it * 128;
            async_copy_b128(lds_h + idx * 16, gh + idx * 4);
        }
    }
    // ---- adj tile: int32 -> packed bytes (needs VALU, regular loads) ----
    for (int e4 = t; e4 < (16 * N) / 4; e4 += 128) {
        int e = e4 * 4;
        int r = e >> 10, c = e & (N - 1);
        int4 v = *(const int4*)(adj + (size_t)(i0 + r) * N + c);
        unsigned packed = (unsigned)(v.x > 0) | ((unsigned)(v.y > 0) << 8) |
                          ((unsigned)(v.z > 0) << 16) | ((unsigned)(v.w > 0) << 24);
        *(unsigned*)&s_adj[r][c] = packed;
    }
    if (t < 16) s_f1[t] = f1g[k * N + i0 + t];
    wait_async0();
    __syncthreads();

    const int row = t & 15;   // row within tile
    const int sub = t >> 4;   // 0..7, j-partition
    const float f1v = s_f1[row];

    // ---- pass 1a: row max ----
    float mloc = -3.0e38f;
    for (int j = sub * 128; j < sub * 128 + 128; ++j) {
        float e0 = f1v + s_f2[j];
        e0 = e0 > 0.f ? e0 : ALPHA * e0;          // leaky_relu
        e0 = s_adj[row][j] ? e0 : NEGINF;         // adjacency mask
        mloc = fmaxf(mloc, e0);
    }
    s_red[row][sub] = mloc;
    __syncthreads();
    if (t < 16) {
        float m = s_red[t][0];
#pragma unroll
        for (int u = 1; u < 8; ++u) m = fmaxf(m, s_red[t][u]);
        s_m[t] = m;
    }
    __syncthreads();

    // ---- pass 1b: row sum of exp ----
    const float mrow = s_m[row];
    float sloc = 0.f;
    for (int j = sub * 128; j < sub * 128 + 128; ++j) {
        float e0 = f1v + s_f2[j];
        e0 = e0 > 0.f ? e0 : ALPHA * e0;
        e0 = s_adj[row][j] ? e0 : NEGINF;
        sloc += __expf(e0 - mrow);
    }
    s_red[row][sub] = sloc;
    __syncthreads();
    if (t < 16) {
        float s = 0.f;
#pragma unroll
        for (int u = 0; u < 8; ++u) s += s_red[t][u];
        s_s[t] = s;
    }
    __syncthreads();

    // ---- pass 2: WMMA-accumulated weighted aggregation ----
    // V_WMMA_F32_16X16X4_F32: A = 16x4 exp weights, B = 4x16 (F cols of h).
    // A layout: lane<16 -> M=lane, K={0,1}; lane>=16 -> M=lane-16, K={2,3}.
    const int lane  = t & 31;
    const int wave  = t >> 5;        // 0..3, each covers 256 j's
    const int mA    = lane & 15;     // output row / B column index
    const int khalf = lane >> 4;     // which K-pair this lane holds
    const float f1A   = s_f1[mA];
    const float mrowA = s_m[mA];
    // clamp+mask instead of conditional loads: keeps EXEC uniform, no branches
    const int   colIdx  = (mA < F) ? mA : 0;
    const float colMask = (mA < F) ? 1.f : 0.f;
    v8f c = {};
    const int jbase = wave * 256;
    for (int cb = 0; cb < 64; ++cb) {
        int j0 = jbase + cb * 4 + khalf * 2;
        float e0 = f1A + s_f2[j0];
        e0 = e0 > 0.f ? e0 : ALPHA * e0;
        e0 = s_adj[mA][j0] ? e0 : NEGINF;
        float e1 = f1A + s_f2[j0 + 1];
        e1 = e1 > 0.f ? e1 : ALPHA * e1;
        e1 = s_adj[mA][j0 + 1] ? e1 : NEGINF;
        v2f av, bv;
        av.x = __expf(e0 - mrowA);
        av.y = __expf(e1 - mrowA);
        bv.x = s_h[j0][colIdx] * colMask;         // B: row K=j0,   col n=mA
        bv.y = s_h[j0 + 1][colIdx] * colMask;     // B: row K=j0+1, col n=mA
        c = __builtin_amdgcn_wmma_f32_16x16x4_f32(
                false, av, false, bv, (short)0, c, false, false);
    }
    // C layout: VGPR r -> M=r (lanes 0-15) / M=r+8 (lanes 16-31), N=lane%16
#pragma unroll
    for (int r = 0; r < 8; ++r)
        s_C[wave][khalf ? r + 8 : r][mA] = c[r];
    __syncthreads();

    // ---- combine partials across waves, normalize, ReLU, store ----
    if (t < 16 * F) {
        int m = t / F, f = t % F;
        float tot = s_C[0][m][f] + s_C[1][m][f] + s_C[2][m][f] + s_C[3][m][f];
        float val = tot / s_s[m];
        val = val > 0.f ? val : 0.f;
        out[((size_t)k * N + i0 + m) * F + f] = val;
    }
}

// ---------------------------------------------------------------------------
// Kernel 3: generic f32 WMMA GEMM  C = [relu](A @ B + bias)
// One wave per 16x16 output tile; K stepped by 4 via wmma_f32_16x16x4_f32.
// Clamp+mask guards keep loads unconditional and EXEC all-ones.
// ---------------------------------------------------------------------------
template <bool RELU>
__global__ __launch_bounds__(32) void wmma_gemm_kernel(
    const float* __restrict__ A, const float* __restrict__ B,
    const float* __restrict__ bias, float* __restrict__ C,
    int M, int Nn, int Kd, int lda, int ldb, int ldc)
{
    const int n0 = blockIdx.x * 16;
    const int m0 = blockIdx.y * 16;
    const int lane = threadIdx.x;
    const int mr = lane & 15;
    const int kh = lane >> 4;
    const int am  = m0 + mr;
    const int bn  = n0 + mr;
    const int   amC = am < M ? am : M - 1;
    const float amM = am < M ? 1.f : 0.f;
    const int   bnC = bn < Nn ? bn : Nn - 1;
    const float bnM = bn < Nn ? 1.f : 0.f;
    v8f c = {};
    for (int kk = 0; kk < Kd; kk += 4) {
        int ka = kk + kh * 2;
        int   ka0 = ka < Kd ? ka : Kd - 1;
        float km0 = ka < Kd ? 1.f : 0.f;
        int   ka1 = ka + 1 < Kd ? ka + 1 : Kd - 1;
        float km1 = ka + 1 < Kd ? 1.f : 0.f;
        v2f av, bv;
        av.x = A[(size_t)amC * lda + ka0] * (amM * km0);
        av.y = A[(size_t)amC * lda + ka1] * (amM * km1);
        bv.x = B[(size_t)ka0 * ldb + bnC] * (bnM * km0);
        bv.y = B[(size_t)ka1 * ldb + bnC] * (bnM * km1);
        c = __builtin_amdgcn_wmma_f32_16x16x4_f32(
                false, av, false, bv, (short)0, c, false, false);
    }
#pragma unroll
    for (int r = 0; r < 8; ++r) {
        int m = m0 + (kh ? r + 8 : r);
        int n = n0 + mr;
        if (m < M && n < Nn) {
            float v = c[r] + (bias ? bias[n] : 0.f);
            if (RELU) v = v > 0.f ? v : 0.f;
            C[(size_t)m * ldc + n] = v;
        }
    }
}

// ---------------------------------------------------------------------------
extern "C" void kernel_launch(void* const* d_in, const int* in_sizes, int n_in,
                              void* d_out, int out_size, void* d_ws, size_t ws_size,
                              hipStream_t stream) {
    (void)in_sizes; (void)n_in; (void)out_size; (void)ws_size;
    const float* X     = (const float*)d_in[0];
    const int*   adj   = (const int*)  d_in[1];
    const float* W1    = (const float*)d_in[2];
    const float* a1    = (const float*)d_in[3];
    const float* W2    = (const float*)d_in[4];
    const float* a2    = (const float*)d_in[5];
    const float* fc1_w = (const float*)d_in[6];
    const float* fc1_b = (const float*)d_in[7];
    const float* out_w = (const float*)d_in[8];
    const float* out_b = (const float*)d_in[9];

    const int Kb = 64, N = 1024;
    const int KN = Kb * N;

    float* ws   = (float*)d_ws;
    float* h1   = ws;                 // [64,1024,4]
    float* f1_1 = h1 + KN * 4;        // [64,1024]
    float* f2_1 = f1_1 + KN;
    float* g1   = f2_1 + KN;          // [64,1024,4]
    float* h2   = g1 + KN * 4;        // [64,1024,8]
    float* f1_2 = h2 + KN * 8;
    float* f2_2 = f1_2 + KN;
    float* g2   = f2_2 + KN;          // [64,1024,8] == x flat [64, 8192]
    float* y1   = g2 + KN * 8;        // [64, 54]

    // Layer 1
    hf_kernel<16, 4><<<KN / 256, 256, 0, stream>>>(X, W1, a1, h1, f1_1, f2_1, KN);
    gat_attn_kernel<4><<<dim3(N / 16, Kb), 128, 0, stream>>>(h1, f1_1, f2_1, adj, g1);
    // Layer 2
    hf_kernel<4, 8><<<KN / 256, 256, 0, stream>>>(g1, W2, a2, h2, f1_2, f2_2, KN);
    gat_attn_kernel<8><<<dim3(N / 16, Kb), 128, 0, stream>>>(h2, f1_2, f2_2, adj, g2);
    // Head: y1 = relu(x @ fc1_w + fc1_b)   [64,8192]@[8192,54]
    wmma_gemm_kernel<true><<<dim3(4, 4), 32, 0, stream>>>(
        g2, fc1_w, fc1_b, y1, 64, 54, 8192, 8192, 54, 54);
    // out = y1 @ out_w + out_b             [64,54]@[54,1024]
    wmma_gemm_kernel<false><<<dim3(N / 16, 4), 32, 0, stream>>>(
        y1, out_w, out_b, (float*)d_out, 64, 1024, 54, 54, 1024, 1024);
}